// GAT_77841987273048
// MI455X (gfx1250) — compile-verified
//
#include <hip/hip_runtime.h>
#include <math.h>

// ---------------------------------------------------------------------------
// GAT layer for MI455X (gfx1250, wave32).
//   h = x @ W                      -> WMMA f32 16x16x4
//   a_src/a_dst = h @ att vectors  -> per-thread dot
//   segment softmax over edges     -> sortable-uint atomicMax + atomicAdd f32
//   scatter alpha*h[src] -> dst    -> one thread per (edge, channel)
//   out = relu(acc+bias) @ W_lin + b_lin -> WMMA f32 16x16x4 (fused relu/bias)
// ---------------------------------------------------------------------------

typedef __attribute__((ext_vector_type(2))) float v2f;
typedef __attribute__((ext_vector_type(8))) float v8f;

#define IN_C  128
#define HID_C 64
#define OUT_C 32
#define NEG_SLOPE 0.2f

// ---------------- WMMA fragment helpers (ISA 7.12.2 layouts) ----------------
// A (16x4 f32):  lanes 0-15 -> M = lane, VGPR0=K0, VGPR1=K1;
//                lanes 16-31 -> M = lane-16, VGPR0=K2, VGPR1=K3.
// B (4x16 f32):  mirrored: lane%16 = N, VGPR pair = K = (lane/16)*2, +1.
// C/D (16x16):   VGPR r: lanes 0-15 -> M=r, N=lane; lanes 16-31 -> M=8+r.

// ---------------- Kernel 0: workspace init ----------------------------------
__global__ void gat_init_kernel(unsigned* __restrict__ mmax_u,
                                float* __restrict__ denom,
                                float* __restrict__ accum, int n) {
  int i = blockIdx.x * blockDim.x + threadIdx.x;
  if (i < n) {
    mmax_u[i] = 0u;      // encoded -inf (below enc of any finite float)
    denom[i]  = 0.0f;
  }
  if (i < n * HID_C) accum[i] = 0.0f;
}

// ---------------- Kernel 1: h = x @ W  (WMMA f32 16x16x4) -------------------
__global__ void gat_gemm_h_kernel(const float* __restrict__ x,
                                  const float* __restrict__ W,
                                  float* __restrict__ h, int n) {
  int wave = (blockIdx.x * blockDim.x + threadIdx.x) >> 5;
  int lane = threadIdx.x & 31;
  int m0   = wave * 16;
  if (m0 >= n) return;                       // uniform per wave

  int mrow = m0 + (lane & 15);
  if (mrow >= n) mrow = n - 1;               // clamp tail loads
  int kq   = (lane >> 4) << 1;               // 0 or 2
  int ncol = lane & 15;

  v8f acc0 = {}, acc1 = {}, acc2 = {}, acc3 = {};
  const float* xr = x + (size_t)mrow * IN_C;

  for (int k0 = 0; k0 < IN_C; k0 += 4) {
    v2f a;
    a.x = xr[k0 + kq];
    a.y = xr[k0 + kq + 1];
    const float* W0 = W + (size_t)(k0 + kq) * HID_C;
    const float* W1 = W0 + HID_C;
    v2f b0, b1, b2, b3;
    b0.x = W0[ncol];       b0.y = W1[ncol];
    b1.x = W0[16 + ncol];  b1.y = W1[16 + ncol];
    b2.x = W0[32 + ncol];  b2.y = W1[32 + ncol];
    b3.x = W0[48 + ncol];  b3.y = W1[48 + ncol];
    acc0 = __builtin_amdgcn_wmma_f32_16x16x4_f32(false, a, false, b0, (short)0, acc0, false, false);
    acc1 = __builtin_amdgcn_wmma_f32_16x16x4_f32(false, a, false, b1, (short)0, acc1, false, false);
    acc2 = __builtin_amdgcn_wmma_f32_16x16x4_f32(false, a, false, b2, (short)0, acc2, false, false);
    acc3 = __builtin_amdgcn_wmma_f32_16x16x4_f32(false, a, false, b3, (short)0, acc3, false, false);
  }

  int rbase = m0 + ((lane >> 4) << 3);
  for (int r = 0; r < 8; ++r) {
    int row = rbase + r;
    if (row < n) {
      float* hr = h + (size_t)row * HID_C;
      hr[ncol]      = acc0[r];
      hr[16 + ncol] = acc1[r];
      hr[32 + ncol] = acc2[r];
      hr[48 + ncol] = acc3[r];
    }
  }
}

// ---------------- Kernel 2: attention dot products --------------------------
__global__ void gat_dots_kernel(const float* __restrict__ h,
                                const float* __restrict__ att_src,
                                const float* __restrict__ att_dst,
                                float* __restrict__ a_src,
                                float* __restrict__ a_dst, int n) {
  int i = blockIdx.x * blockDim.x + threadIdx.x;
  if (i >= n) return;
  const float* hr = h + (size_t)i * HID_C;
  float s = 0.f, d = 0.f;
  for (int c = 0; c < HID_C; ++c) {
    float v = hr[c];
    s = fmaf(v, att_src[c], s);
    d = fmaf(v, att_dst[c], d);
  }
  a_src[i] = s;
  a_dst[i] = d;
}

// ---------------- Kernel 3: edge logits + segment max -----------------------
__device__ __forceinline__ unsigned enc_f32(float f) {
  unsigned u = __float_as_uint(f);
  return (u & 0x80000000u) ? ~u : (u | 0x80000000u);
}
__device__ __forceinline__ float dec_f32(unsigned u) {
  unsigned b = (u & 0x80000000u) ? (u ^ 0x80000000u) : ~u;
  return __uint_as_float(b);
}

__global__ void gat_logit_max_kernel(const int* __restrict__ ei,
                                     const float* __restrict__ a_src,
                                     const float* __restrict__ a_dst,
                                     float* __restrict__ evals,
                                     unsigned* __restrict__ mmax_u,
                                     int n, int e) {
  int i = blockIdx.x * blockDim.x + threadIdx.x;
  int tot = e + n;
  if (i >= tot) return;
  int s, d;
  if (i < e) { s = ei[i]; d = ei[e + i]; }
  else       { s = d = i - e; }              // self loops
  float v = a_src[s] + a_dst[d];
  v = (v >= 0.f) ? v : NEG_SLOPE * v;        // leaky relu
  evals[i] = v;
  atomicMax(mmax_u + d, enc_f32(v));         // GLOBAL_ATOMIC_MAX_U32
}

// ---------------- Kernel 4: exp + segment sum -------------------------------
__global__ void gat_exp_sum_kernel(const int* __restrict__ ei,
                                   const unsigned* __restrict__ mmax_u,
                                   float* __restrict__ evals,
                                   float* __restrict__ denom,
                                   int n, int e) {
  int i = blockIdx.x * blockDim.x + threadIdx.x;
  int tot = e + n;
  if (i >= tot) return;
  int d = (i < e) ? ei[e + i] : (i - e);
  float m = dec_f32(mmax_u[d]);
  float p = __expf(evals[i] - m);
  evals[i] = p;                              // reuse buffer for numerator
  atomicAdd(denom + d, p);                   // GLOBAL_ATOMIC_ADD_F32
}

// ---------------- Kernel 5: scatter alpha * h[src] into accum ---------------
__global__ void gat_scatter_kernel(const int* __restrict__ ei,
                                   const float* __restrict__ h,
                                   const float* __restrict__ evals,
                                   const float* __restrict__ denom,
                                   float* __restrict__ accum,
                                   int n, int e) {
  int i = blockIdx.x * blockDim.x + threadIdx.x;   // (e+n)*64 threads
  int tot = e + n;
  int idx = i >> 6;                                 // edge id
  int c   = i & 63;                                 // channel
  if (idx >= tot) return;
  int s, d;
  if (idx < e) { s = ei[idx]; d = ei[e + idx]; }
  else         { s = d = idx - e; }
  float alpha = evals[idx] / (denom[d] + 1e-16f);   // uniform over 64 threads
  float msg   = alpha * h[(size_t)s * HID_C + c];   // coalesced gather via L2
  atomicAdd(accum + (size_t)d * HID_C + c, msg);
}

// ---------------- Kernel 6: out = relu(acc+bias) @ W_lin + b_lin (WMMA) -----
__global__ void gat_gemm_out_kernel(const float* __restrict__ accum,
                                    const float* __restrict__ bias_conv,
                                    const float* __restrict__ W_lin,
                                    const float* __restrict__ b_lin,
                                    float* __restrict__ out, int n) {
  int wave = (blockIdx.x * blockDim.x + threadIdx.x) >> 5;
  int lane = threadIdx.x & 31;
  int m0   = wave * 16;
  if (m0 >= n) return;

  int mrow = m0 + (lane & 15);
  if (mrow >= n) mrow = n - 1;
  int kq   = (lane >> 4) << 1;
  int ncol = lane & 15;

  v8f acc0 = {}, acc1 = {};
  const float* ar = accum + (size_t)mrow * HID_C;

  for (int k0 = 0; k0 < HID_C; k0 += 4) {
    int ka = k0 + kq, kb = k0 + kq + 1;
    v2f a;
    a.x = fmaxf(ar[ka] + bias_conv[ka], 0.f);       // fused bias + relu
    a.y = fmaxf(ar[kb] + bias_conv[kb], 0.f);
    const float* W0 = W_lin + (size_t)ka * OUT_C;
    const float* W1 = W0 + OUT_C;
    v2f b0, b1;
    b0.x = W0[ncol];       b0.y = W1[ncol];
    b1.x = W0[16 + ncol];  b1.y = W1[16 + ncol];
    acc0 = __builtin_amdgcn_wmma_f32_16x16x4_f32(false, a, false, b0, (short)0, acc0, false, false);
    acc1 = __builtin_amdgcn_wmma_f32_16x16x4_f32(false, a, false, b1, (short)0, acc1, false, false);
  }

  int rbase = m0 + ((lane >> 4) << 3);
  float bl0 = b_lin[ncol];
  float bl1 = b_lin[16 + ncol];
  for (int r = 0; r < 8; ++r) {
    int row = rbase + r;
    if (row < n) {
      float* orow = out + (size_t)row * OUT_C;
      orow[ncol]      = acc0[r] + bl0;
      orow[16 + ncol] = acc1[r] + bl1;
    }
  }
}

// ---------------------------------------------------------------------------
extern "C" void kernel_launch(void* const* d_in, const int* in_sizes, int n_in,
                              void* d_out, int out_size, void* d_ws, size_t ws_size,
                              hipStream_t stream) {
  const float* x         = (const float*)d_in[0];
  const int*   ei        = (const int*)d_in[1];
  const float* W         = (const float*)d_in[2];
  const float* att_src   = (const float*)d_in[3];
  const float* att_dst   = (const float*)d_in[4];
  const float* bias_conv = (const float*)d_in[5];
  const float* W_lin     = (const float*)d_in[6];
  const float* b_lin     = (const float*)d_in[7];
  float*       out       = (float*)d_out;

  const int n = in_sizes[0] / IN_C;     // 50000
  const int e = in_sizes[1] / 2;        // 800000
  const int tot = e + n;

  // workspace carve-out
  char* ws = (char*)d_ws;
  size_t off = 0;
  float*    h      = (float*)(ws + off);    off += (size_t)n * HID_C * sizeof(float);
  float*    a_src  = (float*)(ws + off);    off += (size_t)n * sizeof(float);
  float*    a_dst  = (float*)(ws + off);    off += (size_t)n * sizeof(float);
  unsigned* mmax_u = (unsigned*)(ws + off); off += (size_t)n * sizeof(unsigned);
  float*    denom  = (float*)(ws + off);    off += (size_t)n * sizeof(float);
  float*    evals  = (float*)(ws + off);    off += (size_t)tot * sizeof(float);
  float*    accum  = (float*)(ws + off);    off += (size_t)n * HID_C * sizeof(float);

  const int TB = 256;

  // 0) init accumulators
  gat_init_kernel<<<(n * HID_C + TB - 1) / TB, TB, 0, stream>>>(mmax_u, denom, accum, n);

  // 1) h = x @ W  (one wave per 16 rows)
  {
    int waves   = (n + 15) / 16;
    int threads = waves * 32;
    gat_gemm_h_kernel<<<(threads + TB - 1) / TB, TB, 0, stream>>>(x, W, h, n);
  }

  // 2) attention dots
  gat_dots_kernel<<<(n + TB - 1) / TB, TB, 0, stream>>>(h, att_src, att_dst, a_src, a_dst, n);

  // 3) logits + segment max
  gat_logit_max_kernel<<<(tot + TB - 1) / TB, TB, 0, stream>>>(ei, a_src, a_dst, evals, mmax_u, n, e);

  // 4) exp + segment sum
  gat_exp_sum_kernel<<<(tot + TB - 1) / TB, TB, 0, stream>>>(ei, mmax_u, evals, denom, n, e);

  // 5) message scatter (one thread per edge-channel)
  {
    long long work = (long long)tot * HID_C;
    gat_scatter_kernel<<<(int)((work + TB - 1) / TB), TB, 0, stream>>>(ei, h, evals, denom, accum, n, e);
  }

  // 6) final GEMM with fused relu/bias
  {
    int waves   = (n + 15) / 16;
    int threads = waves * 32;
    gat_gemm_out_kernel<<<(threads + TB - 1) / TB, TB, 0, stream>>>(accum, bias_conv, W_lin, b_lin, out, n);
  }
}